// ClusterMergeNet_3848290697821
// MI455X (gfx1250) — compile-verified
//
#include <hip/hip_runtime.h>
#include <math.h>

typedef __attribute__((ext_vector_type(2))) float v2f;
typedef __attribute__((ext_vector_type(8))) float v8f;

#define NB 4      // batches
#define NN 512    // points per batch
#define DD 128    // feature dim
#define HH 128    // hidden dim
#define BN (NB * NN)   // 2048 flattened rows

// ---------------------------------------------------------------------------
// Stage 1: three fused fp32 WMMA GEMMs over the flattened [2048 x 128] X:
//   mat 0: A'  = X @ W1c[0:128,:]   + b1c          (pair "i" half, bias folded)
//   mat 1: Bm  = X @ W1c[128:256,:]                (pair "j" half)
//   mat 2: Hs  = relu(X @ W1s + b1s)               (self-sim hidden)
// One wave (32 threads) per 16-row tile; 8 f32 accumulators span all 128 H.
// Uses V_WMMA_F32_16X16X4_F32, K advanced 4 per step.
// ---------------------------------------------------------------------------
__global__ __launch_bounds__(32)
void cm_stage1_gemm(const float* __restrict__ X,
                    const float* __restrict__ W1c, const float* __restrict__ b1c,
                    const float* __restrict__ W1s, const float* __restrict__ b1s,
                    float* __restrict__ Ap, float* __restrict__ Bm,
                    float* __restrict__ Hs)
{
    const int rt    = blockIdx.x;        // 0..127 : 16-row tile of the 2048 rows
    const int mat   = blockIdx.y;        // 0..2   : which weight matrix
    const int lane  = threadIdx.x;       // 0..31
    const int laneM = lane & 15;         // row within tile (A) / col within tile (B,D)
    const int laneK = lane >> 4;         // 0 or 1 : K-pair select / row-half select

    const float* W;
    const float* bias;
    float* out;
    bool relu;
    if (mat == 0)      { W = W1c;           bias = b1c;     out = Ap; relu = false; }
    else if (mat == 1) { W = W1c + DD * HH; bias = nullptr; out = Bm; relu = false; }
    else               { W = W1s;           bias = b1s;     out = Hs; relu = true;  }

    const int rowBase = rt * 16;

    v8f acc[8] = {};   // 8 tiles of 16x16 f32 -> rows [rowBase,rowBase+16) x all 128 H

    for (int kk = 0; kk < DD; kk += 4) {
        // A fragment (16x4 f32): lane (m, kpair) holds X[row m][kk+2*kpair + {0,1}]
        const v2f a = *(const v2f*)(X + (size_t)(rowBase + laneM) * DD + kk + 2 * laneK);
        const int k0 = kk + 2 * laneK;
#pragma unroll
        for (int ht = 0; ht < 8; ++ht) {
            // B fragment (4x16 f32): lane holds W[k0][hcol], W[k0+1][hcol]
            v2f bf;
            bf.x = W[(size_t)k0 * HH + ht * 16 + laneM];
            bf.y = W[(size_t)(k0 + 1) * HH + ht * 16 + laneM];
            acc[ht] = __builtin_amdgcn_wmma_f32_16x16x4_f32(
                false, a, false, bf, (short)0, acc[ht], false, false);
        }
    }

    // Epilogue: bias (+relu) and store.  D layout: vgpr r, lanes 0-15 -> M=r,
    // lanes 16-31 -> M=8+r; N = laneM in both halves.
#pragma unroll
    for (int ht = 0; ht < 8; ++ht) {
        const float bv = bias ? bias[ht * 16 + laneM] : 0.0f;
#pragma unroll
        for (int r = 0; r < 8; ++r) {
            float v = acc[ht][r] + bv;
            if (relu) v = fmaxf(v, 0.0f);
            out[(size_t)(rowBase + r + 8 * laneK) * HH + ht * 16 + laneM] = v;
        }
    }
}

// ---------------------------------------------------------------------------
// Stage 2: diagonal.  One wave per row: sim[b,n,n] = sigmoid(Hs[row]·W2s + b2s)
// ---------------------------------------------------------------------------
__global__ __launch_bounds__(256)
void cm_diag(const float* __restrict__ Hs, const float* __restrict__ W2s,
             const float* __restrict__ b2s, float* __restrict__ out)
{
    const int wave = threadIdx.x >> 5;
    const int lane = threadIdx.x & 31;
    const int row  = blockIdx.x * 8 + wave;          // 0..2047
    const int b    = row >> 9;
    const int n    = row & (NN - 1);

    float p = 0.0f;
#pragma unroll
    for (int h = lane; h < HH; h += 32)
        p = fmaf(Hs[(size_t)row * HH + h], W2s[h], p);
#pragma unroll
    for (int off = 16; off > 0; off >>= 1)
        p += __shfl_xor(p, off, 32);

    if (lane == 0) {
        const float t = p + b2s[0];
        out[(size_t)b * NN * NN + (size_t)n * NN + n] = 1.0f / (1.0f + __expf(-t));
    }
}

// ---------------------------------------------------------------------------
// Stage 3: pair interaction.  Block = 32x32 (i,j) tile for one batch.
//   sim = sigmoid( sum_h relu(A'[i][h] + Bm[j][h]) * W2c[h] + b2c )
// A'/Bm tiles staged in LDS (pitch 132 dwords -> conflict-free b128 reads).
// Writes both (i,j) and (j,i); with tiles tj>=ti and the i<j guard this
// covers every off-diagonal element exactly once.
// ---------------------------------------------------------------------------
__global__ __launch_bounds__(1024)
void cm_pairs(const float* __restrict__ Ap, const float* __restrict__ Bm,
              const float* __restrict__ W2c, const float* __restrict__ b2c,
              float* __restrict__ out)
{
    __shared__ float sA[32][132];
    __shared__ float sB[32][132];

    const int b = blockIdx.y;

    // decode blockIdx.x in [0,136) -> (ti_tile, tj_tile) with tj>=ti, 16 tiles
    int tp = blockIdx.x, tit = 0;
    while (tp >= 16 - tit) { tp -= 16 - tit; ++tit; }
    const int tjt = tit + tp;

    const int tid = threadIdx.x;
    const int w = tid >> 5;     // 0..31  (row within tile for loads; j for compute)
    const int l = tid & 31;     // 0..31  (float4 column for loads; i for compute)

    // cooperative tile loads: each thread one float4
    {
        const float4 av = *(const float4*)(Ap + (size_t)(b * NN + tit * 32 + w) * HH + l * 4);
        *(float4*)&sA[w][l * 4] = av;
        const float4 bv = *(const float4*)(Bm + (size_t)(b * NN + tjt * 32 + w) * HH + l * 4);
        *(float4*)&sB[w][l * 4] = bv;
    }
    __syncthreads();

    const int i = tit * 32 + l;   // lane-varying  -> sA read is conflict-free
    const int j = tjt * 32 + w;   // wave-uniform  -> sB read is a broadcast

    if (i < j) {
        float t = b2c[0];
#pragma unroll 4
        for (int h = 0; h < HH; h += 4) {
            const float4 a  = *(const float4*)&sA[l][h];
            const float4 bb = *(const float4*)&sB[w][h];
            const float4 wv = *(const float4*)&W2c[h];   // uniform -> scalar loads
            t = fmaf(fmaxf(a.x + bb.x, 0.0f), wv.x, t);
            t = fmaf(fmaxf(a.y + bb.y, 0.0f), wv.y, t);
            t = fmaf(fmaxf(a.z + bb.z, 0.0f), wv.z, t);
            t = fmaf(fmaxf(a.w + bb.w, 0.0f), wv.w, t);
        }
        const float s = 1.0f / (1.0f + __expf(-t));
        out[(size_t)b * NN * NN + (size_t)i * NN + j] = s;
        out[(size_t)b * NN * NN + (size_t)j * NN + i] = s;
    }
}

// ---------------------------------------------------------------------------
extern "C" void kernel_launch(void* const* d_in, const int* in_sizes, int n_in,
                              void* d_out, int out_size, void* d_ws, size_t ws_size,
                              hipStream_t stream)
{
    const float* X   = (const float*)d_in[0];
    const float* W1c = (const float*)d_in[1];
    const float* b1c = (const float*)d_in[2];
    const float* W2c = (const float*)d_in[3];
    const float* b2c = (const float*)d_in[4];
    const float* W1s = (const float*)d_in[5];
    const float* b1s = (const float*)d_in[6];
    const float* W2s = (const float*)d_in[7];
    const float* b2s = (const float*)d_in[8];
    float* out = (float*)d_out;

    // workspace: A' | Bm | Hs, each 2048x128 f32 (1 MiB each, 3 MiB total)
    float* Ap = (float*)d_ws;
    float* Bmw = Ap + (size_t)BN * HH;
    float* Hs = Bmw + (size_t)BN * HH;

    cm_stage1_gemm<<<dim3(BN / 16, 3), 32, 0, stream>>>(X, W1c, b1c, W1s, b1s,
                                                        Ap, Bmw, Hs);
    cm_diag<<<dim3(BN / 8), 256, 0, stream>>>(Hs, W2s, b2s, out);
    cm_pairs<<<dim3(136, NB), 1024, 0, stream>>>(Ap, Bmw, W2c, b2c, out);
}